// AnnularPatchEmbed_790273983103
// MI455X (gfx1250) — compile-verified
//
#include <hip/hip_runtime.h>
#include <hip/hip_bf16.h>

typedef float v2f __attribute__((ext_vector_type(2)));
typedef float v8f __attribute__((ext_vector_type(8)));

#define IMGD  224
#define PPIX  50176        // 224*224
#define NRING 16
#define TOKD  256
#define OUTD  192
#define NBAT  64
#define SEGSTRIDE 5120     // per-ring index slots (multiple of 64, > max ring pixel count ~4772)
#define BK    64
#define SPLITK 8

// ---------------------------------------------------------------------------
// Kernel 1: build per-ring pixel index lists from the mask planes.
// One wave per ring, sequential ballot-compaction => deterministic ordering.
// Segments padded to a multiple of BK with -1 sentinels.
// ---------------------------------------------------------------------------
__global__ __launch_bounds__(32) void build_perm_kernel(const float* __restrict__ masks,
                                                        int* __restrict__ perm,
                                                        int* __restrict__ kcnt) {
    const int r    = blockIdx.x;
    const int lane = threadIdx.x;
    const float* mrow = masks + r * PPIX;
    int* seg = perm + r * SEGSTRIDE;
    int cnt = 0;
    for (int p0 = 0; p0 < PPIX; p0 += 32) {
        const int p = p0 + lane;
        const bool in = (mrow[p] != 0.0f);
        const unsigned bal = __builtin_amdgcn_ballot_w32(in);
        const int pos = cnt + __builtin_popcount(bal & ((1u << lane) - 1u));
        if (in) seg[pos] = p;
        cnt += __builtin_popcount(bal);
    }
    const int kc = (cnt + BK - 1) & ~(BK - 1);
    for (int i = cnt + lane; i < kc; i += 32) seg[i] = -1;
    if (lane == 0) kcnt[r] = kc;
}

// ---------------------------------------------------------------------------
// Kernel 2: segmented GEMM. For ring r: tokens[b, r, d] = sum over ring pixels
// of x[b,p] * W[d,p].
//
// Block = 256 threads (8 waves) computes ALL 64 batches (M) x 128 dims (N) for
// one (ring, n-half, k-split). Wave w owns N tile [w*16, w*16+16) and four
// independent 16x16 f32 accumulators (one per 16-batch M tile) -> four
// independent WMMA chains. W and x are gathered ONCE per ring across splits.
//
// Padding entries (pix == -1) use a clamped address (pixel 0, a corner pixel,
// valid memory) and zero only the A operand: 0 * finite = 0, so B may load
// unconditionally. All staging is branchless.
//
// Assumed f32 WMMA fragment layout (mirrors ISA 16x4 A layout):
//   A: lane l -> row M=l&15; VGPRs hold K = kbase + (l<16 ? {0,1} : {2,3})
//   B: lane l -> col N=l&15; VGPRs hold same K pairs
//   C: vgpr j -> M = j + (l>=16 ? 8 : 0), N = l&15
// LDS tiles use k-stride 68 => conflict-free reads/writes.
// ---------------------------------------------------------------------------
__global__ __launch_bounds__(256) void ring_gemm_kernel(const float* __restrict__ x,
                                                        const float* __restrict__ W,
                                                        const int*   __restrict__ perm,
                                                        const int*   __restrict__ kcnt,
                                                        float*       __restrict__ partial) {
    __shared__ float A_lds[64 * 68];    // 17.4 KB
    __shared__ float B_lds[128 * 68];   // 34.8 KB

    const int r  = blockIdx.x & 15;
    const int n2 = blockIdx.x >> 4;     // 0..1
    const int s  = blockIdx.y;          // 0..SPLITK-1

    const int tid  = threadIdx.x;
    const int lane = tid & 31;
    const int w    = tid >> 5;

    const int ktiles = kcnt[r] >> 6;
    const int* seg   = perm + r * SEGSTRIDE;

    const int mn  = lane & 15;
    const int sel = (lane >> 4) << 1;   // 0 for lanes 0-15, 2 for lanes 16-31

    v8f c0 = {}, c1 = {}, c2 = {}, c3 = {};

    for (int kt = s; kt < ktiles; kt += SPLITK) {
        const int kb = kt * BK;

        // ---- stage A tile: A_lds[m][kk] = x[m][perm[kb+kk]] (0 for padding)
        {
            const int kk   = tid & 63;
            const int mb   = tid >> 6;             // 0..3
            const int pix  = seg[kb + kk];
            const int pa   = (pix < 0) ? 0 : pix;  // clamped valid address
            #pragma unroll
            for (int mi = 0; mi < 16; ++mi) {
                const int m = mb + mi * 4;
                float v = x[m * PPIX + pa];        // unconditional load
                v = (pix < 0) ? 0.0f : v;          // branchless zero for pad
                A_lds[m * 68 + kk] = v;
            }
        }
        // ---- stage B tile: B_lds[dd][kk] = W[n2*128+dd][perm[kb+kk]]
        {
            const int q0  = seg[kb + lane];
            const int q1  = seg[kb + 32 + lane];
            const int pa0 = (q0 < 0) ? 0 : q0;
            const int pa1 = (q1 < 0) ? 0 : q1;
            #pragma unroll
            for (int i = 0; i < 16; ++i) {
                const int dd = w * 16 + i;
                const long drow = (long)(n2 * 128 + dd) * PPIX;
                B_lds[dd * 68 + lane]      = W[drow + pa0];   // pad rows are
                B_lds[dd * 68 + 32 + lane] = W[drow + pa1];   // killed by A==0
            }
        }
        __syncthreads();

        // ---- 16 WMMA K-steps x 4 M tiles over this 64-wide chunk
        const float* Brow = &B_lds[(w * 16 + mn) * 68];
        const float* Arow = &A_lds[mn * 68];
        #pragma unroll
        for (int kq = 0; kq < BK; kq += 4) {
            v2f b;
            b.x = Brow[kq + sel];  b.y = Brow[kq + sel + 1];
            v2f a0, a1, a2, a3;
            a0.x = Arow[0 * 16 * 68 + kq + sel];  a0.y = Arow[0 * 16 * 68 + kq + sel + 1];
            a1.x = Arow[1 * 16 * 68 + kq + sel];  a1.y = Arow[1 * 16 * 68 + kq + sel + 1];
            a2.x = Arow[2 * 16 * 68 + kq + sel];  a2.y = Arow[2 * 16 * 68 + kq + sel + 1];
            a3.x = Arow[3 * 16 * 68 + kq + sel];  a3.y = Arow[3 * 16 * 68 + kq + sel + 1];
            c0 = __builtin_amdgcn_wmma_f32_16x16x4_f32(false, a0, false, b, (short)0, c0, false, false);
            c1 = __builtin_amdgcn_wmma_f32_16x16x4_f32(false, a1, false, b, (short)0, c1, false, false);
            c2 = __builtin_amdgcn_wmma_f32_16x16x4_f32(false, a2, false, b, (short)0, c2, false, false);
            c3 = __builtin_amdgcn_wmma_f32_16x16x4_f32(false, a3, false, b, (short)0, c3, false, false);
        }
        __syncthreads();
    }

    // ---- store the four C tiles into split-K partial buffer partial[s][row][col]
    float* pb = partial + (size_t)s * (1024 * TOKD);
    const int col = n2 * 128 + w * 16 + mn;
    #pragma unroll
    for (int j = 0; j < 8; ++j) {
        const int M = j + ((lane >> 4) << 3);
        pb[((0 * 16 + M) * NRING + r) * TOKD + col] = c0[j];
        pb[((1 * 16 + M) * NRING + r) * TOKD + col] = c1[j];
        pb[((2 * 16 + M) * NRING + r) * TOKD + col] = c2[j];
        pb[((3 * 16 + M) * NRING + r) * TOKD + col] = c3[j];
    }
}

// ---------------------------------------------------------------------------
// Kernel 3: fixed-order split-K reduction -> tokens[1024][256]
// ---------------------------------------------------------------------------
__global__ __launch_bounds__(256) void reduce_kernel(const float* __restrict__ partial,
                                                     float* __restrict__ tokens) {
    const int i = blockIdx.x * 256 + threadIdx.x;
    const int n = 1024 * TOKD;
    if (i < n) {
        float v = partial[i];
        #pragma unroll
        for (int s = 1; s < SPLITK; ++s) v += partial[(size_t)s * n + i];
        tokens[i] = v;
    }
}

// ---------------------------------------------------------------------------
// Kernel 4: out[row, dout] = tokens[row,:] @ fc_w[dout,:] + fc_b[dout]
// One wave per 16x16 output tile; K=256 via 64 f32 WMMA steps.
// ---------------------------------------------------------------------------
__global__ __launch_bounds__(32) void fc_gemm_kernel(const float* __restrict__ tokens,
                                                     const float* __restrict__ fc_w,
                                                     const float* __restrict__ fc_b,
                                                     float* __restrict__ out) {
    const int mt   = blockIdx.x;   // 0..63
    const int nt   = blockIdx.y;   // 0..11
    const int lane = threadIdx.x;
    const int mn   = lane & 15;
    const int sel  = (lane >> 4) << 1;

    const float* arow = tokens + (mt * 16 + mn) * TOKD;
    const float* brow = fc_w   + (nt * 16 + mn) * TOKD;  // B[k][n] = fc_w[n][k]

    v8f c = {};
    #pragma unroll 8
    for (int k = 0; k < TOKD; k += 4) {
        v2f a, b;
        a.x = arow[k + sel];  a.y = arow[k + sel + 1];
        b.x = brow[k + sel];  b.y = brow[k + sel + 1];
        c = __builtin_amdgcn_wmma_f32_16x16x4_f32(false, a, false, b,
                                                  (short)0, c, false, false);
    }
    #pragma unroll
    for (int j = 0; j < 8; ++j) {
        const int M   = j + ((lane >> 4) << 3);
        const int col = nt * 16 + mn;
        out[(mt * 16 + M) * OUTD + col] = c[j] + fc_b[col];
    }
}

// ---------------------------------------------------------------------------
extern "C" void kernel_launch(void* const* d_in, const int* in_sizes, int n_in,
                              void* d_out, int out_size, void* d_ws, size_t ws_size,
                              hipStream_t stream) {
    const float* x     = (const float*)d_in[0];  // [64, 224, 224]
    const float* tw    = (const float*)d_in[1];  // [256, 1, 224, 224]
    const float* fc_w  = (const float*)d_in[2];  // [192, 256]
    const float* fc_b  = (const float*)d_in[3];  // [192]
    const float* masks = (const float*)d_in[4];  // [16, 224, 224]
    float* out = (float*)d_out;                  // [64, 16, 192]

    // Workspace layout (~9.8 MB)
    char* ws = (char*)d_ws;
    int*   perm    = (int*)ws;                              // 16*5120*4       = 327680 B
    int*   kcnt    = (int*)(ws + 327680);                   // 64 B
    float* partial = (float*)(ws + 328192);                 // 8*1024*256*4    = 8 MiB
    float* tokens  = (float*)(ws + 328192 + 8388608);       // 1024*256*4      = 1 MiB

    build_perm_kernel<<<dim3(NRING), dim3(32), 0, stream>>>(masks, perm, kcnt);
    ring_gemm_kernel<<<dim3(32, SPLITK), dim3(256), 0, stream>>>(x, tw, perm, kcnt, partial);
    reduce_kernel<<<dim3(1024), dim3(256), 0, stream>>>(partial, tokens);
    fc_gemm_kernel<<<dim3(64, 12), dim3(32), 0, stream>>>(tokens, fc_w, fc_b, out);
}